// AttentionSortNet_43104291782971
// MI455X (gfx1250) — compile-verified
//
#include <hip/hip_runtime.h>
#include <hip/hip_bf16.h>

typedef __attribute__((ext_vector_type(2))) float v2f;
typedef __attribute__((ext_vector_type(8))) float v8f;

#define BH_N 32
#define SEQ 8192
#define DIM_N 64
#define BUCKETS 64
#define BSZ 128
#define EPS_F 1e-6f
#define TEMP_F 0.7f
#define SINK_IT 8

// ---------------------------------------------------------------------------
// Phase 1: bucket means.  grid = 4096 blocks (2 tensors x 32 bh x 64 buckets),
// 256 threads.  Each block reduces a 128x64 tile (32 KB) to 64 column means.
// float4 loads, fully coalesced; dominates runtime -> pure HBM stream.
// ---------------------------------------------------------------------------
__global__ __launch_bounds__(256) void bucket_mean_kernel(
    const float* __restrict__ q, const float* __restrict__ k,
    float* __restrict__ sq, float* __restrict__ sk) {
  int b = blockIdx.x;                 // 0..4095
  bool is_k = b >= 2048;
  int bb = is_k ? (b - 2048) : b;
  int bh = bb >> 6;
  int bucket = bb & 63;
  const float* src = (is_k ? k : q) +
                     ((size_t)bh * SEQ + (size_t)bucket * BSZ) * DIM_N;
  float* dst = (is_k ? sk : sq) + ((size_t)bh * BUCKETS + bucket) * DIM_N;

  int t = threadIdx.x;
  int c4 = t & 15;                    // column-group (4 floats), constant per thread
  const float4* src4 = (const float4*)src;  // 2048 float4 per tile
  float4 acc = make_float4(0.f, 0.f, 0.f, 0.f);
#pragma unroll
  for (int j = 0; j < 8; ++j) {
    float4 v = src4[t + 256 * j];     // rows t/16 + 16j, cols 4*c4..4*c4+3
    acc.x += v.x; acc.y += v.y; acc.z += v.z; acc.w += v.w;
  }
  __shared__ float4 part[16][16];     // [row-group][col-group]
  part[t >> 4][c4] = acc;
  __syncthreads();
  if (t < 64) {
    int cg = t >> 2, comp = t & 3;
    float s = 0.f;
#pragma unroll
    for (int g = 0; g < 16; ++g) {
      float4 p = part[g][cg];
      s += (comp == 0) ? p.x : (comp == 1) ? p.y : (comp == 2) ? p.z : p.w;
    }
    dst[t] = s * (1.0f / 128.0f);
  }
}

// ---------------------------------------------------------------------------
// Phase 2: per-bh 64x64 GEMM via V_WMMA_F32_16X16X4_F32 + gumbel-sinkhorn.
// grid = 32 blocks, 256 threads (8 waves; 2 16x16 tiles per wave).
// r kept in LDS with stride 65 -> conflict-free row AND column passes.
// ---------------------------------------------------------------------------
__global__ __launch_bounds__(256) void sortnet_kernel(
    const float* __restrict__ sq, const float* __restrict__ sk,
    const float* __restrict__ gu, float* __restrict__ out) {
  int bh = blockIdx.x;
  const float* SQ = sq + (size_t)bh * 64 * 64;
  const float* SK = sk + (size_t)bh * 64 * 64;
  const float* GU = gu + (size_t)bh * 64 * 64;
  float* OUT = out + (size_t)bh * 64 * 64;

  __shared__ float r[64 * 65];
  __shared__ float lrow[64];
  __shared__ float lcol[64];

  int tid = threadIdx.x;
  int wave = tid >> 5;
  int lane = tid & 31;
  int lm = lane & 15;   // row (A) / col (B) within tile
  int hi = lane >> 4;   // K-half select

  // R = SQ * SK^T * dim^-0.5, then r = (log(relu(R)+eps) + gumbel) / T
#pragma unroll
  for (int tt = 0; tt < 2; ++tt) {
    int tile = wave * 2 + tt;         // 0..15
    int ti = tile >> 2;
    int tj = tile & 3;
    v8f c = {};
    const float* abase = SQ + (ti * 16 + lm) * 64 + 2 * hi;
    const float* bbase = SK + (tj * 16 + lm) * 64 + 2 * hi;
#pragma unroll
    for (int kk = 0; kk < 64; kk += 4) {
      v2f a = *(const v2f*)(abase + kk);
      v2f b = *(const v2f*)(bbase + kk);
      // (neg_a, A, neg_b, B, c_mod, C, reuse_a, reuse_b)
      c = __builtin_amdgcn_wmma_f32_16x16x4_f32(false, a, false, b,
                                                (short)0, c, false, false);
    }
    // C layout: VGPR v -> row v (lanes 0-15) / v+8 (lanes 16-31), lane -> col
#pragma unroll
    for (int v = 0; v < 8; ++v) {
      int row = ti * 16 + v + 8 * hi;
      int col = tj * 16 + lm;
      float Rv = c[v] * 0.125f;                       // * 64^-0.5
      float u = GU[row * 64 + col];
      float gl = -logf(-logf(u + EPS_F) + EPS_F);     // gumbel noise
      r[row * 65 + col] = (logf(fmaxf(Rv, 0.f) + EPS_F) + gl) / TEMP_F;
    }
  }
  __syncthreads();

  for (int it = 0; it < SINK_IT; ++it) {
    if (tid < 64) {                   // row logsumexp (axis=2)
      const float* rp = &r[tid * 65];
      float mx = rp[0];
      for (int j = 1; j < 64; ++j) mx = fmaxf(mx, rp[j]);
      float s = 0.f;
      for (int j = 0; j < 64; ++j) s += expf(rp[j] - mx);
      lrow[tid] = mx + logf(s);
    }
    __syncthreads();
    if (tid < 64) {                   // col logsumexp (axis=1) of row-normed r
      float mx = r[tid] - lrow[0];
      for (int i = 1; i < 64; ++i) mx = fmaxf(mx, r[i * 65 + tid] - lrow[i]);
      float s = 0.f;
      for (int i = 0; i < 64; ++i) s += expf(r[i * 65 + tid] - lrow[i] - mx);
      lcol[tid] = mx + logf(s);
    }
    __syncthreads();
#pragma unroll
    for (int e = 0; e < 16; ++e) {    // fused row+col subtraction
      int idx = tid + 256 * e;
      int i = idx >> 6, j = idx & 63;
      r[i * 65 + j] -= lrow[i] + lcol[j];
    }
    __syncthreads();
  }

#pragma unroll
  for (int e = 0; e < 16; ++e) {
    int idx = tid + 256 * e;
    int i = idx >> 6, j = idx & 63;
    OUT[idx] = expf(r[i * 65 + j]);
  }
}

extern "C" void kernel_launch(void* const* d_in, const int* in_sizes, int n_in,
                              void* d_out, int out_size, void* d_ws, size_t ws_size,
                              hipStream_t stream) {
  (void)in_sizes; (void)n_in; (void)out_size; (void)ws_size;
  const float* q  = (const float*)d_in[0];
  const float* k  = (const float*)d_in[1];
  const float* gu = (const float*)d_in[2];
  float* out = (float*)d_out;
  float* sq = (float*)d_ws;                    // 32*64*64 floats
  float* sk = sq + (size_t)BH_N * BUCKETS * DIM_N;  // +512 KB

  bucket_mean_kernel<<<4096, 256, 0, stream>>>(q, k, sq, sk);
  sortnet_kernel<<<BH_N, 256, 0, stream>>>(sq, sk, gu, out);
}